// LLaMATransformer_57080115364609
// MI455X (gfx1250) — compile-verified
//
#include <hip/hip_runtime.h>
#include <math.h>
#include <stdint.h>

typedef __attribute__((ext_vector_type(2))) float v2f;
typedef __attribute__((ext_vector_type(8))) float v8f;
typedef __attribute__((ext_vector_type(4))) unsigned int v4u;
typedef __attribute__((ext_vector_type(8))) int v8i;
typedef __attribute__((ext_vector_type(4))) int v4i;

#define D_MODEL 1024
#define NHEADS  16
#define HD      64
#define BATCH   8
#define NLAYER  4
#define SEQ     2048
#define SM1     2047                 // cached positions per layer (S-1)
#define FFDIM   2816
#define VOCAB   32000
#define KV_SLOTS (NLAYER*2*SM1)      // 16376
#define EPS 1e-5f

// ---------------- embedding gather ----------------
__global__ void k_embed(const int* __restrict__ tokens, const float* __restrict__ embed,
                        float* __restrict__ h) {
  int b = blockIdx.x;
  int t = tokens[b];
  const float* src = embed + (size_t)t * D_MODEL;
  for (int d = threadIdx.x; d < D_MODEL; d += blockDim.x)
    h[b * D_MODEL + d] = src[d];
}

// ---------------- RMSNorm (one block per batch row) ----------------
__global__ void k_rmsnorm(const float* __restrict__ x, const float* __restrict__ w,
                          float* __restrict__ out) {
  __shared__ float red[256];
  int b = blockIdx.x;
  const float* xb = x + b * D_MODEL;
  float s = 0.f;
  for (int d = threadIdx.x; d < D_MODEL; d += 256) { float v = xb[d]; s += v * v; }
  red[threadIdx.x] = s;
  __syncthreads();
  for (int st = 128; st > 0; st >>= 1) {
    if (threadIdx.x < st) red[threadIdx.x] += red[threadIdx.x + st];
    __syncthreads();
  }
  float scale = rsqrtf(red[0] / (float)D_MODEL + EPS);
  for (int d = threadIdx.x; d < D_MODEL; d += 256)
    out[b * D_MODEL + d] = xb[d] * scale * w[d];
}

// ---------------- TDM: async-load one 8 x 64 f32 activation panel into LDS ----------
// D# per CDNA5 ISA §8: group0 = {count=1, lds_addr, global_addr, type=2}
//                      group1 = {data_size=4B, tensor_dim0=K, tensor_dim1=8,
//                                tile_dim0=64, tile_dim1=8, dim0_stride=K}
// Tracked by TENSORcnt; EXEC ignored (one issue per wave).
// 6-arg builtin form (clang-23): (v4u, v8i, v4i, v4i, v8i, i32 cpol).
__device__ __forceinline__ void tdm_load_panel(const float* gsrc, unsigned ldsAddr, int K) {
  unsigned long long ga = (unsigned long long)(uintptr_t)gsrc;
  v4u g0;
  g0.x = 1u;                                      // count=1 (valid user descriptor)
  g0.y = ldsAddr;                                 // lds_addr (bytes, from real &xbuf)
  g0.z = (unsigned)(ga & 0xFFFFFFFFu);            // global_addr[31:0]
  g0.w = (unsigned)((ga >> 32) & 0x01FFFFFFu)     // global_addr[56:32]
         | (2u << 30);                            // type=2 ("image")
  v8i g1;
  g1[0] = 0x00020000;                             // data_size=2 (4 bytes), no multicast
  g1[1] = (K & 0xFFFF) << 16;                     // tensor_dim0[15:0] = K
  g1[2] = (8 << 16) | ((K >> 16) & 0xFFFF);       // tensor_dim1=8 | tensor_dim0[31:16]
  g1[3] = (64 << 16);                             // tile_dim0 = 64
  g1[4] = 8;                                      // tile_dim1 = 8
  g1[5] = K;                                      // tensor_dim0_stride = K
  g1[6] = 0;
  g1[7] = 0;
  v4i z4 = {0, 0, 0, 0};                          // 2-D tensor: groups 2/3 unused
  v8i z8 = {0, 0, 0, 0, 0, 0, 0, 0};              // unused extra group (zero-filled)
  __builtin_amdgcn_tensor_load_to_lds(g0, g1, z4, z4, z8, 0);
}

// ---------------- skinny GEMM via V_WMMA_F32_16X16X4_F32 + TDM staging --------------
// Y[8][N] = X[8][K] @ W[K][N] (+ addend[8][N] if non-null).
// One wave owns one 16-col tile. Activations double-buffered in LDS by the TDM:
// wave 0 issues panel p+1, s_wait_tensorcnt(1) => panel p landed, barrier publishes,
// 8 waves run 16 back-to-back WMMAs on panel p while panel p+1 streams in.
// The TDM lds_addr is derived from the real address of xbuf (generic->LDS pointers
// carry the LDS byte offset in their low 32 bits) and the pointer is escaped so the
// compiler cannot treat xbuf as unwritten and fold the A-panel ds_loads.
// Weight rows are walked with a marching pointer (+4N per step, no per-load 64-bit
// multiply) and prefetched WGP-scope 2 panels ahead.
// Grids are sized exactly (N/16 divisible by 8) so all waves reach the barriers.
__global__ void k_gemm_wmma(const float* __restrict__ X, const float* __restrict__ W,
                            const float* __restrict__ addend, float* __restrict__ Y,
                            int K, int N) {
  __shared__ float xbuf[2][8 * 64];
  const int wave = threadIdx.x >> 5;
  const int lane = threadIdx.x & 31;
  const int n0 = (blockIdx.x * 8 + wave) * 16;
  const int half = lane >> 4;
  const int l16  = lane & 15;
  const float rowMask = (l16 < 8) ? 1.0f : 0.0f;
  const float* Wp = W + n0 + l16;                 // this lane's output column
  const int P = K >> 6;                           // K is always a multiple of 64
  const size_t strideN = (size_t)N;

  // LDS byte address of the staging buffer; escape it so LDS is treated as written.
  const unsigned ldsBase = (unsigned)(uintptr_t)(void*)&xbuf[0][0];
  asm volatile("" : : "r"(&xbuf[0][0]) : "memory");

  if (wave == 0) tdm_load_panel(X, ldsBase, K);   // prime panel 0

  v8f c = {};
  for (int p = 0; p < P; ++p) {
    const int k0 = p << 6;
    if (wave == 0) {
      if (p + 1 < P) {
        tdm_load_panel(X + (k0 + 64), ldsBase + (unsigned)(((p + 1) & 1) * (8 * 64 * 4)), K);
        __builtin_amdgcn_s_wait_tensorcnt(1);     // panel p complete, p+1 in flight
      } else {
        __builtin_amdgcn_s_wait_tensorcnt(0);
      }
    }
    __syncthreads();                              // publish panel p to all waves

    const float* xb = &xbuf[p & 1][(l16 & 7) * 64];
    // marching weight pointer: rows (k0 + half*2), advances 4 rows per step
    const float* wr = Wp + ((size_t)k0 + (size_t)(half * 2)) * strideN;
    __builtin_prefetch((const void*)(Wp + (size_t)(k0 + 128) * strideN), 0, 3);
#pragma unroll
    for (int ku = 0; ku < 64; ku += 4) {
      const int kk = ku + half * 2;
      v2f a, bm;
      a.x  = xb[kk]     * rowMask;                // ds_load, rows 8..15 masked to 0
      a.y  = xb[kk + 1] * rowMask;
      bm.x = wr[0];                               // row k0+kk
      bm.y = wr[strideN];                         // row k0+kk+1
      wr += 4 * strideN;
      c = __builtin_amdgcn_wmma_f32_16x16x4_f32(false, a, false, bm, (short)0, c,
                                                false, false);
    }
    __syncthreads();                              // buffer p&1 free for panel p+2
  }

  if (half == 0) {                                // lanes 0-15, VGPR r = row M=r
#pragma unroll
    for (int r = 0; r < 8; ++r) {
      float v = c[r];
      if (addend) v += addend[(size_t)r * N + n0 + l16];
      Y[(size_t)r * N + n0 + l16] = v;
    }
  }
}

// ---------------- RoPE on q and new-k (interleaved pairs) ----------------
__global__ void k_rope(float* __restrict__ q, float* __restrict__ k,
                       const float* __restrict__ fc, const float* __restrict__ fs) {
  int b = blockIdx.x;
  for (int idx = threadIdx.x; idx < NHEADS * (HD / 2); idx += blockDim.x) {
    int hh = idx >> 5, i = idx & 31;
    int base = b * D_MODEL + hh * HD + 2 * i;
    float c = fc[i], s = fs[i];
    float qr = q[base], qi = q[base + 1];
    q[base]     = qr * c - qi * s;
    q[base + 1] = qr * s + qi * c;
    float kr = k[base], ki = k[base + 1];
    k[base]     = kr * c - ki * s;
    k[base + 1] = kr * s + ki * c;
  }
}

// ---------------- cache rotation: old positions 1..SM1-1 -> out slots 0..SM1-2 -------
__global__ void k_copy_cache(const float4* __restrict__ src, float4* __restrict__ dst_kv,
                             int slotBase) {
  const int D4 = D_MODEL / 4;
  size_t idx = (size_t)blockIdx.x * blockDim.x + threadIdx.x;
  const size_t total = (size_t)BATCH * (SM1 - 1) * D4;
  if (idx >= total) return;
  int d4 = (int)(idx % D4);
  size_t t = idx / D4;
  int pos = (int)(t % (SM1 - 1));
  int b   = (int)(t / (SM1 - 1));
  dst_kv[((size_t)b * KV_SLOTS + slotBase + pos) * D4 + d4] =
      src[((size_t)b * SM1 + pos + 1) * D4 + d4];
}

// ---------------- write the new k/v row into the last slot of each section ----------
__global__ void k_write_newkv(const float* __restrict__ kn, const float* __restrict__ vn,
                              float* __restrict__ out_kv, int l) {
  int idx = blockIdx.x * blockDim.x + threadIdx.x;   // 2*B*D threads exactly
  int which = idx / (BATCH * D_MODEL);
  int rem   = idx % (BATCH * D_MODEL);
  int b = rem / D_MODEL;
  int slot = l * 2 * SM1 + which * SM1 + (SM1 - 1);
  out_kv[((size_t)b * KV_SLOTS + slot) * D_MODEL + (rem % D_MODEL)] =
      (which ? vn : kn)[rem];
}

// ---------------- attention scores: q . k / sqrt(hd) + mask ----------------
__global__ void k_scores(const float* __restrict__ q, const float* __restrict__ kc_l,
                         const float* __restrict__ kn, const float* __restrict__ mask,
                         float* __restrict__ scores) {
  int bh = blockIdx.y;             // b*H + h
  int b = bh >> 4, hh = bh & 15;
  int s = blockIdx.x * blockDim.x + threadIdx.x;   // 0..SEQ-1
  const float4* qp = (const float4*)(q + b * D_MODEL + hh * HD);
  const float4* kp = (s < SM1)
      ? (const float4*)(kc_l + ((size_t)b * SM1 + s) * D_MODEL + hh * HD)
      : (const float4*)(kn + b * D_MODEL + hh * HD);
  float acc = 0.f;
#pragma unroll
  for (int j = 0; j < HD / 4; ++j) {
    float4 a = qp[j], k4 = kp[j];
    acc += a.x * k4.x + a.y * k4.y + a.z * k4.z + a.w * k4.w;
  }
  scores[(size_t)bh * SEQ + s] = acc * 0.125f + mask[b * SEQ + s];
}

// ---------------- softmax over SEQ per (b,h) row, in place ----------------
__global__ void k_softmax(float* __restrict__ scores) {
  __shared__ float red[256];
  float* p = scores + (size_t)blockIdx.x * SEQ;
  float m = -1e30f;
  for (int i = threadIdx.x; i < SEQ; i += 256) m = fmaxf(m, p[i]);
  red[threadIdx.x] = m;
  __syncthreads();
  for (int st = 128; st > 0; st >>= 1) {
    if (threadIdx.x < st) red[threadIdx.x] = fmaxf(red[threadIdx.x], red[threadIdx.x + st]);
    __syncthreads();
  }
  m = red[0];
  __syncthreads();
  float s = 0.f;
  for (int i = threadIdx.x; i < SEQ; i += 256) {
    float e = __expf(p[i] - m);
    p[i] = e;
    s += e;
  }
  red[threadIdx.x] = s;
  __syncthreads();
  for (int st = 128; st > 0; st >>= 1) {
    if (threadIdx.x < st) red[threadIdx.x] += red[threadIdx.x + st];
    __syncthreads();
  }
  float inv = 1.f / red[0];
  for (int i = threadIdx.x; i < SEQ; i += 256) p[i] *= inv;
}

// ---------------- o = P @ V  (64 threads = one head dim each) ----------------
__global__ void k_attnout(const float* __restrict__ scores, const float* __restrict__ vc_l,
                          const float* __restrict__ vn, float* __restrict__ o) {
  int bh = blockIdx.x, b = bh >> 4, hh = bh & 15;
  const float* p = scores + (size_t)bh * SEQ;
  int d = threadIdx.x;   // 0..63
  const float* vp = vc_l + (size_t)b * SM1 * D_MODEL + hh * HD + d;
  float acc = 0.f;
  for (int s = 0; s < SM1; ++s) acc += p[s] * vp[(size_t)s * D_MODEL];
  acc += p[SM1] * vn[b * D_MODEL + hh * HD + d];
  o[b * D_MODEL + hh * HD + d] = acc;
}

// ---------------- SwiGLU gate ----------------
__global__ void k_silu_mul(const float* __restrict__ g1, const float* __restrict__ g3,
                           float* __restrict__ hh) {
  int idx = blockIdx.x * blockDim.x + threadIdx.x;
  if (idx >= BATCH * FFDIM) return;
  float a = g1[idx];
  hh[idx] = (a / (1.f + __expf(-a))) * g3[idx];
}

extern "C" void kernel_launch(void* const* d_in, const int* in_sizes, int n_in,
                              void* d_out, int out_size, void* d_ws, size_t ws_size,
                              hipStream_t stream) {
  const int*   tokens = (const int*)d_in[0];
  const float* fcos   = (const float*)d_in[1];
  const float* fsin   = (const float*)d_in[2];
  const float* mask   = (const float*)d_in[3];
  const float* kcache = (const float*)d_in[4];
  const float* vcache = (const float*)d_in[5];
  const float* embed  = (const float*)d_in[6];
  const float* wq     = (const float*)d_in[7];
  const float* wk     = (const float*)d_in[8];
  const float* wv     = (const float*)d_in[9];
  const float* wo     = (const float*)d_in[10];
  const float* attn_w = (const float*)d_in[11];
  const float* ffn_w  = (const float*)d_in[12];
  const float* w1     = (const float*)d_in[13];
  const float* w2     = (const float*)d_in[14];
  const float* w3     = (const float*)d_in[15];
  const float* fin_w  = (const float*)d_in[16];
  const float* w_out  = (const float*)d_in[17];

  float* logits = (float*)d_out;
  float* out_kv = logits + (size_t)BATCH * VOCAB;

  float* ws = (float*)d_ws;
  float* h  = ws; ws += BATCH * D_MODEL;
  float* xn = ws; ws += BATCH * D_MODEL;
  float* q  = ws; ws += BATCH * D_MODEL;
  float* kn = ws; ws += BATCH * D_MODEL;
  float* vn = ws; ws += BATCH * D_MODEL;
  float* o  = ws; ws += BATCH * D_MODEL;
  float* g1 = ws; ws += BATCH * FFDIM;
  float* g3 = ws; ws += BATCH * FFDIM;
  float* hh = ws; ws += BATCH * FFDIM;
  float* sc = ws; ws += (size_t)BATCH * NHEADS * SEQ;

  auto gemm = [&](const float* X, const float* W, const float* add, float* Y,
                  int K, int N) {
    int blocks = (N / 16) / 8;   // exact for N = 1024 / 2816 / 32000
    k_gemm_wmma<<<blocks, 256, 0, stream>>>(X, W, add, Y, K, N);
  };

  k_embed<<<BATCH, 256, 0, stream>>>(tokens, embed, h);

  const size_t layerW = (size_t)D_MODEL * D_MODEL;
  const size_t layerC = (size_t)BATCH * SM1 * D_MODEL;
  const int copyBlocks =
      (int)(((size_t)BATCH * (SM1 - 1) * (D_MODEL / 4) + 255) / 256);

  for (int l = 0; l < NLAYER; ++l) {
    k_rmsnorm<<<BATCH, 256, 0, stream>>>(h, attn_w + l * D_MODEL, xn);
    gemm(xn, wq + l * layerW, nullptr, q,  D_MODEL, D_MODEL);
    gemm(xn, wk + l * layerW, nullptr, kn, D_MODEL, D_MODEL);
    gemm(xn, wv + l * layerW, nullptr, vn, D_MODEL, D_MODEL);
    k_rope<<<BATCH, 256, 0, stream>>>(q, kn, fcos, fsin);

    k_copy_cache<<<copyBlocks, 256, 0, stream>>>(
        (const float4*)(kcache + l * layerC), (float4*)out_kv, l * 2 * SM1);
    k_copy_cache<<<copyBlocks, 256, 0, stream>>>(
        (const float4*)(vcache + l * layerC), (float4*)out_kv, l * 2 * SM1 + SM1);
    k_write_newkv<<<(2 * BATCH * D_MODEL) / 256, 256, 0, stream>>>(kn, vn, out_kv, l);

    k_scores<<<dim3(SEQ / 256, BATCH * NHEADS), 256, 0, stream>>>(
        q, kcache + l * layerC, kn, mask, sc);
    k_softmax<<<BATCH * NHEADS, 256, 0, stream>>>(sc);
    k_attnout<<<BATCH * NHEADS, HD, 0, stream>>>(sc, vcache + l * layerC, vn, o);
    gemm(o, wo + l * layerW, h, h, D_MODEL, D_MODEL);

    k_rmsnorm<<<BATCH, 256, 0, stream>>>(h, ffn_w + l * D_MODEL, xn);
    gemm(xn, w1 + (size_t)l * D_MODEL * FFDIM, nullptr, g1, D_MODEL, FFDIM);
    gemm(xn, w3 + (size_t)l * D_MODEL * FFDIM, nullptr, g3, D_MODEL, FFDIM);
    k_silu_mul<<<(BATCH * FFDIM + 255) / 256, 256, 0, stream>>>(g1, g3, hh);
    gemm(hh, w2 + (size_t)l * FFDIM * D_MODEL, h, h, FFDIM, D_MODEL);
  }

  k_rmsnorm<<<BATCH, 256, 0, stream>>>(h, fin_w, xn);
  gemm(xn, w_out, nullptr, logits, D_MODEL, VOCAB);

  (void)in_sizes; (void)n_in; (void)out_size; (void)ws_size;
}